// DecNP_49331994362184
// MI455X (gfx1250) — compile-verified
//
#include <hip/hip_runtime.h>
#include <math.h>

typedef __attribute__((ext_vector_type(2))) float v2f;
typedef __attribute__((ext_vector_type(8))) float v8f;

#define GAMMA  0.85f
#define KNBR   8
#define MDIR   20
#define DFEAT  768
#define CHUNK  512
#define BLK    256

// ---------------------------------------------------------------------------
// Kernel A: WMMA-based squared-distance tiles -> per-point top-8 -> direction
// weights. Each wave handles 32 rows of xyz1 via two 16x16x4 F32 WMMA tiles.
// Distance tile = A(-2*xyz1) x B(xyz2^T) + C(s2+d2)  -- C carries the norms.
// ---------------------------------------------------------------------------
__global__ __launch_bounds__(BLK) void topk_weight_kernel(
    const float* __restrict__ xyz1,   // [N,3] fine points
    const float* __restrict__ xyz2,   // [S,3] coarse points
    const float* __restrict__ perc,   // [S,MDIR]
    const float* __restrict__ dirs,   // [S,MDIR,3]
    int S,
    int*   __restrict__ idx_out,      // [N,KNBR]
    float* __restrict__ wgt_out,      // [N,KNBR]
    float* __restrict__ wsum_out)     // [N]
{
    __shared__ float sx[CHUNK], sy[CHUNK], sz[CHUNK], sd2[CHUNK];
    __shared__ float s2sh[BLK];
    __shared__ float distS[8][32][16];

    const int tid  = threadIdx.x;
    const int wid  = tid >> 5;
    const int lane = tid & 31;
    const int l16  = lane & 15;
    const int half = (lane >> 4) & 1;          // 0: holds K=0,1 ; 1: holds K=2,3
    const int blockBase = blockIdx.x * BLK;
    const int rowBase   = blockBase + wid * 32;
    const int n = blockBase + tid;             // point owned by this thread

    // Own point coords + squared norm (shared so lanes can build C fragments).
    float px = xyz1[n*3+0], py = xyz1[n*3+1], pz = xyz1[n*3+2];
    s2sh[tid] = px*px + py*py + pz*pz;

    // A fragments (16x4, f32): lanes 0-15 hold K=0,1; lanes 16-31 hold K=2,3.
    // Loop-invariant: load once. A' = -2 * xyz1 so WMMA yields -2*dot.
    const int m0 = rowBase + l16;
    const int m1 = rowBase + 16 + l16;
    v2f a0, a1;
    if (half == 0) {
        a0.x = -2.0f * xyz1[m0*3+0]; a0.y = -2.0f * xyz1[m0*3+1];
        a1.x = -2.0f * xyz1[m1*3+0]; a1.y = -2.0f * xyz1[m1*3+1];
    } else {
        a0.x = -2.0f * xyz1[m0*3+2]; a0.y = 0.0f;   // K=3 pad
        a1.x = -2.0f * xyz1[m1*3+2]; a1.y = 0.0f;
    }
    __syncthreads();

    // Base C fragments: row-norm part (s2). Column-norm (d2) added per tile.
    v8f c0b, c1b;
    #pragma unroll
    for (int v = 0; v < 8; ++v) {
        c0b[v] = s2sh[wid*32 + v + half*8];
        c1b[v] = s2sh[wid*32 + 16 + v + half*8];
    }

    // Running sorted top-8 (ascending squared distance) -- fully unrolled so
    // it stays in VGPRs.
    float bd[KNBR]; int bi[KNBR];
    #pragma unroll
    for (int k = 0; k < KNBR; ++k) { bd[k] = INFINITY; bi[k] = 0; }

    for (int cs = 0; cs < S; cs += CHUNK) {
        __syncthreads();
        for (int t = tid; t < CHUNK; t += BLK) {
            int s = cs + t;
            float bx = xyz2[s*3+0], by = xyz2[s*3+1], bz = xyz2[s*3+2];
            sx[t] = bx; sy[t] = by; sz[t] = bz;
            sd2[t] = bx*bx + by*by + bz*bz;
        }
        __syncthreads();

        for (int cb = 0; cb < CHUNK; cb += 16) {
            const int ccol = cb + l16;
            // B fragment (4x16): VGPR0 = rows K=0 (lanes 0-15) / K=2 (16-31),
            // VGPR1 = rows K=1 / K=3(=0 pad).
            v2f b;
            if (half == 0) { b.x = sx[ccol]; b.y = sy[ccol]; }
            else           { b.x = sz[ccol]; b.y = 0.0f; }
            const float d2c = sd2[ccol];
            v8f c0, c1;
            #pragma unroll
            for (int v = 0; v < 8; ++v) { c0[v] = c0b[v] + d2c; c1[v] = c1b[v] + d2c; }

            // D = s2 + d2 - 2*dot  == squared distances, directly.
            v8f d0 = __builtin_amdgcn_wmma_f32_16x16x4_f32(
                false, a0, false, b, (short)0, c0, false, false);
            v8f d1 = __builtin_amdgcn_wmma_f32_16x16x4_f32(
                false, a1, false, b, (short)0, c1, false, false);

            // Transpose tile through per-wave LDS so each lane owns one row.
            #pragma unroll
            for (int v = 0; v < 8; ++v) {
                distS[wid][v + half*8][l16]      = d0[v];
                distS[wid][16 + v + half*8][l16] = d1[v];
            }
            asm volatile("s_wait_dscnt 0" ::: "memory");  // same-wave LDS RAW

            const float* drow = &distS[wid][lane][0];
            #pragma unroll
            for (int j = 0; j < 16; ++j) {
                float dv = drow[j];
                int   gc = cs + cb + j;
                if (dv < bd[KNBR-1]) {          // strict < : earlier idx wins ties
                    bd[KNBR-1] = dv; bi[KNBR-1] = gc;
                    #pragma unroll
                    for (int q = KNBR-1; q > 0; --q) {
                        if (bd[q] < bd[q-1]) {
                            float td = bd[q]; bd[q] = bd[q-1]; bd[q-1] = td;
                            int   ti = bi[q]; bi[q] = bi[q-1]; bi[q-1] = ti;
                        }
                    }
                }
            }
        }
    }

    // --- direction-gated weighting (scalar, L2-resident data) ---
    float wk[KNBR]; float wsum = 0.0f;
    #pragma unroll
    for (int k = 0; k < KNBR; ++k) {
        int s = bi[k];
        float nx = xyz2[s*3+0], ny = xyz2[s*3+1], nz = xyz2[s*3+2];
        float vx = nx - px, vy = ny - py, vz = nz - pz;
        float dd  = sqrtf(vx*vx + vy*vy + vz*vz);
        float inv = 1.0f / (dd + 1e-8f);
        vx *= inv; vy *= inv; vz *= inv;
        float w = 0.0f;
        for (int m = 0; m < MDIR; ++m) {
            const float* dp = dirs + ((size_t)s*MDIR + m)*3;
            float dx = dp[0], dy = dp[1], dz = dp[2];
            float dn = sqrtf(dx*dx + dy*dy + dz*dz) + 1e-8f;
            float simm = (dx*vx + dy*vy + dz*vz) / dn;
            if (fabsf(simm) > GAMMA) w += perc[s*MDIR + m];
        }
        wk[k] = w; wsum += w;
    }
    wsum += 1e-8f;
    float dr8[KNBR]; float nrm = 0.0f;
    #pragma unroll
    for (int k = 0; k < KNBR; ++k) {
        dr8[k] = wk[k] / wsum + 1e-6f + 1e-10f;   // de_k_weight_norm + 1e-10
        nrm += dr8[k];
    }
    nrm += 1e-8f;
    float invn = 1.0f / nrm;
    #pragma unroll
    for (int k = 0; k < KNBR; ++k) {
        idx_out[n*KNBR+k] = bi[k];
        wgt_out[n*KNBR+k] = dr8[k] * invn;
    }
    wsum_out[n] = wsum;
}

// ---------------------------------------------------------------------------
// Kernel B: deterministic reduction  scal = 1e-8 + c * mean(wsum)
// ---------------------------------------------------------------------------
__global__ __launch_bounds__(BLK) void reduce_scal_kernel(
    const float* __restrict__ wsum, int N, float c, float* __restrict__ scal_out)
{
    __shared__ float sh[BLK];
    float acc = 0.0f;
    for (int i = threadIdx.x; i < N; i += BLK) acc += wsum[i];
    sh[threadIdx.x] = acc;
    __syncthreads();
    for (int o = BLK/2; o > 0; o >>= 1) {
        if (threadIdx.x < o) sh[threadIdx.x] += sh[threadIdx.x + o];
        __syncthreads();
    }
    if (threadIdx.x == 0) scal_out[0] = 1e-8f + c * (sh[0] / (float)N);
}

// ---------------------------------------------------------------------------
// Kernel C: gather-interpolate D=768 features, combine, L2-normalize.
// One block per point; 3 elements per thread.
// ---------------------------------------------------------------------------
__global__ __launch_bounds__(BLK) void interp_norm_kernel(
    const float* __restrict__ P1,     // [N,D] fine features
    const float* __restrict__ P2,     // [S,D] coarse features
    const int*   __restrict__ idx,    // [N,KNBR]
    const float* __restrict__ wgt,    // [N,KNBR]
    const float* __restrict__ wsum,   // [N]
    const float* __restrict__ scal_p, // [1]
    float* __restrict__ out)          // [N,D]
{
    const int n = blockIdx.x;
    __shared__ float sh[BLK];
    __shared__ int   sidx[KNBR];
    __shared__ float swgt[KNBR];
    if (threadIdx.x < KNBR) {
        sidx[threadIdx.x] = idx[n*KNBR + threadIdx.x];
        swgt[threadIdx.x] = wgt[n*KNBR + threadIdx.x];
    }
    __syncthreads();
    const float ws   = wsum[n];
    const float scal = scal_p[0];

    float vals[DFEAT/BLK];
    float ssq = 0.0f;
    #pragma unroll
    for (int r = 0; r < DFEAT/BLK; ++r) {
        int d = threadIdx.x + r*BLK;
        float acc = 0.0f;
        #pragma unroll
        for (int k = 0; k < KNBR; ++k)
            acc += swgt[k] * P2[(size_t)sidx[k]*DFEAT + d];
        float v = acc * ws + scal * P1[(size_t)n*DFEAT + d];
        vals[r] = v;
        ssq += v * v;
    }
    sh[threadIdx.x] = ssq;
    __syncthreads();
    for (int o = BLK/2; o > 0; o >>= 1) {
        if (threadIdx.x < o) sh[threadIdx.x] += sh[threadIdx.x + o];
        __syncthreads();
    }
    float inv = 1.0f / fmaxf(sqrtf(sh[0]), 1e-12f);
    #pragma unroll
    for (int r = 0; r < DFEAT/BLK; ++r)
        out[(size_t)n*DFEAT + threadIdx.x + r*BLK] = vals[r] * inv;
}

// ---------------------------------------------------------------------------
extern "C" void kernel_launch(void* const* d_in, const int* in_sizes, int n_in,
                              void* d_out, int out_size, void* d_ws, size_t ws_size,
                              hipStream_t stream) {
    const float* xyz_c  = (const float*)d_in[0];  // [1,1024,3]
    const float* xyz_m  = (const float*)d_in[1];  // [1,4096,3]
    const float* xyz_f  = (const float*)d_in[2];  // [1,16384,3]
    const float* x_c    = (const float*)d_in[3];  // [1,1024,768]
    const float* x_m    = (const float*)d_in[4];  // [1,4096,768]
    const float* x_f    = (const float*)d_in[5];  // [1,16384,768]
    const float* perc_c = (const float*)d_in[6];  // [1,1024,20]
    const float* dir_c  = (const float*)d_in[7];  // [1,1024,20,3]
    const float* perc_m = (const float*)d_in[8];  // [1,4096,20]
    const float* dir_m  = (const float*)d_in[9];  // [1,4096,20,3]

    const int N0 = 4096,  S0 = 1024;   // stage 0: coarse -> mid
    const int N1 = 16384, S1 = 4096;   // stage 1: mid -> fine

    // Workspace layout
    char* ws = (char*)d_ws;
    size_t off = 0;
    float* x_mid = (float*)(ws + off); off += (size_t)N0 * DFEAT * sizeof(float);
    int*   idxb  = (int*)  (ws + off); off += (size_t)N1 * KNBR * sizeof(int);
    float* wgtb  = (float*)(ws + off); off += (size_t)N1 * KNBR * sizeof(float);
    float* wsb   = (float*)(ws + off); off += (size_t)N1 * sizeof(float);
    float* scal  = (float*)(ws + off);

    // ---- Stage 0: 1024 -> 4096 (N == 4*S -> c = 0.3) ----
    topk_weight_kernel<<<N0/BLK, BLK, 0, stream>>>(
        xyz_m, xyz_c, perc_c, dir_c, S0, idxb, wgtb, wsb);
    reduce_scal_kernel<<<1, BLK, 0, stream>>>(wsb, N0, 0.3f, scal);
    interp_norm_kernel<<<N0, BLK, 0, stream>>>(
        x_m, x_c, idxb, wgtb, wsb, scal, x_mid);

    // ---- Stage 1: 4096 -> 16384 (N == 4*S -> c = 0.3) ----
    topk_weight_kernel<<<N1/BLK, BLK, 0, stream>>>(
        xyz_f, xyz_m, perc_m, dir_m, S1, idxb, wgtb, wsb);
    reduce_scal_kernel<<<1, BLK, 0, stream>>>(wsb, N1, 0.3f, scal);
    interp_norm_kernel<<<N1, BLK, 0, stream>>>(
        x_f, x_mid, idxb, wgtb, wsb, scal, (float*)d_out);
}